// DeformUnfold_37855841747686
// MI455X (gfx1250) — compile-verified
//
#include <hip/hip_runtime.h>

// Deformable unfold (im2col with learned offsets), MI455X / gfx1250.
// x:      [8, 64, 128, 128] fp32
// offset: [8, 18, 128, 128] fp32   (layout [B, K, 2(y,x), Ho, Wo], DG=1)
// out:    [8, 576, 16384]   fp32   (layout [B, C*K, Ho*Wo], channel-major)
//
// Bandwidth-bound gather kernel: no matmul structure -> no WMMA by design.
// gfx1250-specific path used: TH cache hints (NT stores for the 302MB
// write-once output, NT loads for streamed offsets) to keep the 33.5MB
// x tensor resident in the 192MB L2, which serves ~1.2GB of corner gathers.

#define KH 3
#define KW 3
#define KK 9            // KH*KW
#define BATCH 8
#define CHN 64
#define HH 128
#define WW 128
#define HW (HH * WW)    // 16384

__global__ __launch_bounds__(256) void deform_unfold_kernel(
    const float* __restrict__ x,
    const float* __restrict__ offset,
    float* __restrict__ out)
{
    const int p = blockIdx.x * blockDim.x + threadIdx.x;  // spatial pos in Ho*Wo
    if (p >= HW) return;
    const int k = blockIdx.y;                             // kernel tap 0..8
    const int b = blockIdx.z;                             // batch 0..7

    const int wo = p & (WW - 1);
    const int ho = p >> 7;
    const int ky = k / KW;
    const int kx = k - ky * KW;

    // offset channel layout: [B, K, 2, Ho, Wo]; y-plane then x-plane.
    const float* offb = offset + (((size_t)b * KK + k) * 2) * HW + p;
    const float oy = __builtin_nontemporal_load(offb);
    const float ox = __builtin_nontemporal_load(offb + HW);

    // base tap position: ho*STRIDE - PAD + ky*DIL  (STRIDE=DIL=1, PAD=1)
    const float py = (float)(ho - 1 + ky) + oy;
    const float px = (float)(wo - 1 + kx) + ox;

    const float fy = floorf(py);
    const float fx = floorf(px);
    const float ly = py - fy, lx = px - fx;
    const float hy = 1.0f - ly, hx = 1.0f - lx;

    const int y0 = (int)fy, x0 = (int)fx;
    const int y1 = y0 + 1,  x1 = x0 + 1;

    const bool vy0 = (y0 >= 0) & (y0 < HH);
    const bool vy1 = (y1 >= 0) & (y1 < HH);
    const bool vx0 = (x0 >= 0) & (x0 < WW);
    const bool vx1 = (x1 >= 0) & (x1 < WW);

    const int cy0 = min(max(y0, 0), HH - 1);
    const int cy1 = min(max(y1, 0), HH - 1);
    const int cx0 = min(max(x0, 0), WW - 1);
    const int cx1 = min(max(x1, 0), WW - 1);

    const int i00 = cy0 * WW + cx0;
    const int i01 = cy0 * WW + cx1;
    const int i10 = cy1 * WW + cx0;
    const int i11 = cy1 * WW + cx1;

    // zero the weight (not the value) for out-of-bounds corners, matching ref
    const float w00 = hy * hx * (float)(vy0 & vx0);
    const float w01 = hy * lx * (float)(vy0 & vx1);
    const float w10 = ly * hx * (float)(vy1 & vx0);
    const float w11 = ly * lx * (float)(vy1 & vx1);

    const float* xb = x + (size_t)b * CHN * HW;           // x[b, :, :, :]
    // out[((b*C + c)*K + k)*HW + p]; c-stride = K*HW
    float* ob = out + (((size_t)b * CHN) * KK + k) * HW + p;

#pragma unroll 4
    for (int c = 0; c < CHN; ++c) {
        const float* xc = xb + (size_t)c * HW;            // L2-resident gathers
        float v = w00 * xc[i00] + w01 * xc[i01]
                + w10 * xc[i10] + w11 * xc[i11];
        // write-once streaming output: non-temporal (th:TH_STORE_NT),
        // coalesced 32x4B per wave per iteration
        __builtin_nontemporal_store(v, ob + (size_t)c * (KK * HW));
    }
}

extern "C" void kernel_launch(void* const* d_in, const int* in_sizes, int n_in,
                              void* d_out, int out_size, void* d_ws, size_t ws_size,
                              hipStream_t stream) {
    const float* x      = (const float*)d_in[0];  // [8,64,128,128]
    const float* offset = (const float*)d_in[1];  // [8,18,128,128]
    float* out          = (float*)d_out;          // [8,576,16384]

    dim3 block(256, 1, 1);                         // 8 wave32 waves / block
    dim3 grid(HW / 256, KK, BATCH);                // 64 x 9 x 8 = 4608 blocks
    deform_unfold_kernel<<<grid, block, 0, stream>>>(x, offset, out);
}